// DiceLoss_80590766342569
// MI455X (gfx1250) — compile-verified
//
#include <hip/hip_runtime.h>
#include <hip/hip_bf16.h>

// Problem constants (match reference)
#define CC    8            // feat channels
#define CP2   10           // C + 2
#define CWN   169          // dynamic weight vector length
#define KB    32           // objects per image (K)
#define HH    128
#define WW    128
#define HWN   (HH * WW)    // 16384
#define NTILE (HWN / 16)   // 1024 pixel tiles of 16

typedef __attribute__((ext_vector_type(2))) float v2f;
typedef __attribute__((ext_vector_type(8))) float v8f;

// V_WMMA_F32_16X16X4_F32: D(16x16 f32) = A(16x4 f32) * B(4x16 f32) + C
__device__ __forceinline__ v8f wmma_f32x4(v2f a, v2f b, v8f c) {
  return __builtin_amdgcn_wmma_f32_16x16x4_f32(
      /*neg_a=*/false, a, /*neg_b=*/false, b,
      /*c_mod=*/(short)0, c, /*reuse_a=*/false, /*reuse_b=*/false);
}

// One block per (b, object-pair, pixel-split). 256 threads = 8 waves.
// Each wave processes 16-pixel tiles for TWO objects at once:
//   WMMA M rows 0-7 = obj0 channels, rows 8-15 = obj1 channels.
// Writes 3 partial sums (p*t, p*p, t*t) per object per split.
__global__ __launch_bounds__(256)
void dice_main_kernel(const float* __restrict__ seg,      // [B,C,H,W]
                      const float* __restrict__ convw,    // [B,CW,H,W]
                      const int*   __restrict__ mask,     // [B,K]
                      const long long* __restrict__ ind,  // [B,K] int64
                      const float* __restrict__ target,   // [B,K,H,W]
                      float* __restrict__ partial)        // [B*K][2 splits][3]
{
  const int bx     = blockIdx.x;       // 0..255
  const int split  = bx & 1;           // pixel-space half
  const int pairId = bx >> 1;          // 0..127
  const int b      = pairId >> 4;      // 16 pairs per image
  const int k0     = (pairId & 15) * 2;

  __shared__ float wsm[2][CWN];        // gathered weights for both objects
  __shared__ float red[8][2][3];

  const int tid = threadIdx.x;

  const int m0   = mask[b * KB + k0];
  const int m1   = mask[b * KB + k0 + 1];
  const int pos0 = (int)ind[b * KB + k0];
  const int pos1 = (int)ind[b * KB + k0 + 1];

  const int pbase = ((b * KB + k0) * 2 + split) * 3;   // obj0; obj1 at +6

  if ((m0 | m1) == 0) {                // uniform per block
    if (tid < 3)      partial[pbase + tid]     = 0.0f;
    else if (tid < 6) partial[pbase + 3 + tid] = 0.0f;  // +6.. +8
    return;
  }

  // Stage both objects' 169 gathered weights into LDS.
  for (int i = tid; i < CWN; i += 256) {
    wsm[0][i] = convw[((size_t)b * CWN + i) * HWN + pos0];
    wsm[1][i] = convw[((size_t)b * CWN + i) * HWN + pos1];
  }
  __syncthreads();

  const int lane = tid & 31;
  const int wid  = tid >> 5;
  const int col  = lane & 15;          // N (pixel) / M (row) index
  const int half = lane >> 4;          // K sub-pair selector (A/B frags)
  const int aobj = col >> 3;           // object owning A-row `col`
  const int aoc  = col & 7;            // out-channel within that object

  // ---- layer-1 A frags: rows 0-7 obj0 c1w[:, :8], rows 8-15 obj1 ----
  v2f a1[2];
#pragma unroll
  for (int j = 0; j < 2; ++j) {
    const int c0 = 4 * j + 2 * half;                 // seg channel 0..7
    a1[j].x = wsm[aobj][aoc * CP2 + c0];
    a1[j].y = wsm[aobj][aoc * CP2 + c0 + 1];
  }
  // ---- layer-2 A frags: block-diagonal over K=16 ----
  v2f a2[4];
#pragma unroll
  for (int j = 0; j < 4; ++j) {
    const int kk = 4 * j + 2 * half;                 // 0..15 (kk,kk+1 same block)
    const bool valid = ((kk >> 3) == aobj);
    a2[j].x = valid ? wsm[aobj][88 + aoc * 8 + (kk & 7)]       : 0.0f;
    a2[j].y = valid ? wsm[aobj][88 + aoc * 8 + ((kk + 1) & 7)] : 0.0f;
  }

  // ---- per-lane D-side constants: lane's object == half ----
  float c1b[8], c2b[8], c3w[8], cw8[8], cw9[8];
#pragma unroll
  for (int v = 0; v < 8; ++v) {
    c1b[v] = wsm[half][80 + v];
    c2b[v] = wsm[half][152 + v];
    c3w[v] = wsm[half][160 + v];
    cw8[v] = wsm[half][v * CP2 + 8];   // x_rel weight
    cw9[v] = wsm[half][v * CP2 + 9];   // y_rel weight
  }
  const float c3b = wsm[half][168];

  const int   posh = half ? pos1 : pos0;
  const float x0   = (float)(posh & (WW - 1));
  const float y0   = (float)(posh >> 7);
  const float mobj = (half ? m1 : m0) ? 1.0f : 0.0f;

  const float* segb = seg    + (size_t)b * CC * HWN;
  const float* tgt  = target + (size_t)(b * KB + k0 + half) * HWN;

  float s_pt = 0.0f, s_pp = 0.0f, s_tt = 0.0f;

  const int tend = (split + 1) * (NTILE / 2);
  for (int tile = split * (NTILE / 2) + wid; tile < tend; tile += 8) {
    const int px = tile * 16 + col;

    // ---- layer-1 B frags (seg channels 0..7, shared by both objects) ----
    v2f b0, b1;
    const int ch0 = 2 * half;
    b0.x = segb[(size_t)ch0 * HWN + px];
    b0.y = segb[(size_t)(ch0 + 1) * HWN + px];
    const int ch1 = 4 + 2 * half;
    b1.x = segb[(size_t)ch1 * HWN + px];
    b1.y = segb[(size_t)(ch1 + 1) * HWN + px];

    v8f acc = {};
    acc = wmma_f32x4(a1[0], b0, acc);
    acc = wmma_f32x4(a1[1], b1, acc);

    // coordinate channels + bias + relu (per lane: object = half, pixel = col)
    const float xr = ((float)(px & (WW - 1)) - x0) * (1.0f / (float)WW);
    const float yr = ((float)(px >> 7)       - y0) * (1.0f / (float)HH);
    float h1[8];
#pragma unroll
    for (int v = 0; v < 8; ++v) {
      float h = acc[v] + c1b[v];
      h = fmaf(cw8[v], xr, h);
      h = fmaf(cw9[v], yr, h);
      h1[v] = fmaxf(h, 0.0f);
    }

    // ---- layer-2 B frags via single partner-lane exchange (lane ^ 16) ----
    // Each lane sends the 4 channels its partner needs:
    //   half=0 partner (half=1) wants obj0 ch{2,3,6,7};
    //   half=1 partner (half=0) wants obj1 ch{0,1,4,5}.
    const float q0 = half ? h1[0] : h1[2];
    const float q1 = half ? h1[1] : h1[3];
    const float q2 = half ? h1[4] : h1[6];
    const float q3 = half ? h1[5] : h1[7];
    const float r0 = __shfl_xor(q0, 16, 32);
    const float r1 = __shfl_xor(q1, 16, 32);
    const float r2 = __shfl_xor(q2, 16, 32);
    const float r3 = __shfl_xor(q3, 16, 32);

    v2f bb[4];
    bb[0].x = half ? r0 : h1[0];   bb[0].y = half ? r1 : h1[1];
    bb[1].x = half ? r2 : h1[4];   bb[1].y = half ? r3 : h1[5];
    bb[2].x = half ? h1[2] : r0;   bb[2].y = half ? h1[3] : r1;
    bb[3].x = half ? h1[6] : r2;   bb[3].y = half ? h1[7] : r3;

    v8f acc2 = {};
    acc2 = wmma_f32x4(a2[0], bb[0], acc2);
    acc2 = wmma_f32x4(a2[1], bb[1], acc2);
    acc2 = wmma_f32x4(a2[2], bb[2], acc2);
    acc2 = wmma_f32x4(a2[3], bb[3], acc2);

    // ---- layer-3: all 32 lanes useful (object = half, pixel = col) ----
    float s = c3b;
#pragma unroll
    for (int v = 0; v < 8; ++v) {
      const float h2 = fmaxf(acc2[v] + c2b[v], 0.0f);
      s = fmaf(c3w[v], h2, s);
    }
    const float pred = 1.0f / (1.0f + __expf(-s));

    const float t  = tgt[px];
    const float p  = pred * mobj;
    const float tv = t * mobj;
    s_pt = fmaf(p, tv, s_pt);
    s_pp = fmaf(p, p,  s_pp);
    s_tt = fmaf(tv, tv, s_tt);
  }

  // reduce within each 16-lane half (objects stay separate)
#pragma unroll
  for (int off = 8; off >= 1; off >>= 1) {
    s_pt += __shfl_down(s_pt, off, 16);
    s_pp += __shfl_down(s_pp, off, 16);
    s_tt += __shfl_down(s_tt, off, 16);
  }
  if (col == 0) {
    red[wid][half][0] = s_pt;
    red[wid][half][1] = s_pp;
    red[wid][half][2] = s_tt;
  }
  __syncthreads();
  if (tid < 2) {                       // tid = object index
    float pt = 0.0f, pp = 0.0f, tt = 0.0f;
#pragma unroll
    for (int i = 0; i < 8; ++i) {
      pt += red[i][tid][0];
      pp += red[i][tid][1];
      tt += red[i][tid][2];
    }
    const int pb = pbase + tid * 6;    // ((b*KB+k0+tid)*2+split)*3
    partial[pb + 0] = pt;
    partial[pb + 1] = pp;
    partial[pb + 2] = tt;
  }
}

// Deterministic final reduction: per-image dice over 32 objects x 2 splits,
// then mean over B=8 images.
__global__ void dice_final_kernel(const float* __restrict__ partial,
                                  float* __restrict__ out)
{
  __shared__ float lossb[8];
  const int t = threadIdx.x;
  if (t < 8) {
    float pt = 0.0f, pp = 0.0f, tt = 0.0f;
    for (int e = 0; e < KB * 2; ++e) {          // k*2 + split
      const int idx = (t * KB * 2 + e) * 3;
      pt += partial[idx + 0];
      pp += partial[idx + 1];
      tt += partial[idx + 2];
    }
    lossb[t] = 1.0f - (2.0f * pt + 1.0f) / (pp + tt + 1.0f);
  }
  __syncthreads();
  if (t == 0) {
    float s = 0.0f;
#pragma unroll
    for (int i = 0; i < 8; ++i) s += lossb[i];
    out[0] = s * (1.0f / 8.0f);
  }
}

extern "C" void kernel_launch(void* const* d_in, const int* in_sizes, int n_in,
                              void* d_out, int out_size, void* d_ws, size_t ws_size,
                              hipStream_t stream) {
  const float*     seg    = (const float*)d_in[0];      // [8,8,128,128] f32
  const float*     convw  = (const float*)d_in[1];      // [8,169,128,128] f32
  const int*       mask   = (const int*)d_in[2];        // [8,32] i32
  const long long* ind    = (const long long*)d_in[3];  // [8,32] i64
  const float*     target = (const float*)d_in[4];      // [8,32,128,128] f32
  float*           out    = (float*)d_out;              // scalar
  float*           partial = (float*)d_ws;              // 256*2*3 floats

  dice_main_kernel<<<dim3(256), dim3(256), 0, stream>>>(
      seg, convw, mask, ind, target, partial);
  dice_final_kernel<<<dim3(1), dim3(32), 0, stream>>>(partial, out);
}